// tcn_unit_attention_59064390255169
// MI455X (gfx1250) — compile-verified
//
#include <hip/hip_runtime.h>
#include <hip/hip_bf16.h>

typedef __attribute__((ext_vector_type(2))) float v2f;
typedef __attribute__((ext_vector_type(8))) float v8f;

#define NB   16
#define CC   128
#define TT   150
#define VV   25
#define QO   192   // 2*DK + DV
#define NHH  8
#define DKHH 4
#define DVHH 16
#define TPAD 152   // T padded to multiple of 4 for K-loops
#define LSTR 160   // logits rows / row stride (T padded to multiple of 16)

__device__ __forceinline__ v8f wmma4(v2f a, v2f b, v8f c) {
  // D(16x16,f32) = A(16x4,f32) * B(4x16,f32) + C   — full fp32 matrix op
  return __builtin_amdgcn_wmma_f32_16x16x4_f32(false, a, false, b, (short)0, c,
                                               false, false);
}

// Async global->LDS copy (CDNA5 TDM-adjacent path, tracked by ASYNCcnt).
// ldsoff: byte offset of destination within LDS (low 32 bits of generic ptr).
__device__ __forceinline__ void async_g2l_b32(unsigned int ldsoff, const float* g) {
  asm volatile("global_load_async_to_lds_b32 %0, %1, off"
               :: "v"(ldsoff), "v"(g) : "memory");
}
__device__ __forceinline__ void wait_async0() {
  asm volatile("s_wait_asynccnt 0" ::: "memory");
}
__device__ __forceinline__ unsigned int lds_off(const void* p) {
  return (unsigned int)(unsigned long long)p;  // low 32 bits = LDS offset
}

// ---------------------------------------------------------------------------
// Kernel 1: fused data-BN + QKV projection.
// One block per b = n*V + v. B-panel (normalized x, 128 x 152) staged in LDS
// (VGPR path: the copy applies the folded BN affine). A (qkv_w rows)
// preloaded to registers per 16-row tile; 32 K-steps of wmma per output tile.
// q rows (o<32) are pre-scaled by DKH^-0.5 = 0.5 at store so the attention
// kernel's staging is a pure copy (async-friendly).
// ---------------------------------------------------------------------------
extern "C" __global__ __launch_bounds__(256)
void k_qkv(const float* __restrict__ x, const float* __restrict__ qkv_w,
           const float* __restrict__ qkv_b,
           const float* __restrict__ dg, const float* __restrict__ db,
           const float* __restrict__ dm, const float* __restrict__ dvv,
           float* __restrict__ qkv_out) {
  extern __shared__ float sm[];
  float* Bt  = sm;              // CC * TPAD
  float* scl = Bt + CC * TPAD;  // CC
  float* shf = scl + CC;        // CC
  const int b = blockIdx.x;
  const int n = b / VV, v = b % VV;
  const int tid = threadIdx.x;

  if (tid < CC) {
    int cv = tid * VV + v;
    float s = dg[cv] * rsqrtf(dvv[cv] + 1e-5f);
    scl[tid] = s;
    shf[tid] = db[cv] - dm[cv] * s;
  }
  __syncthreads();
  for (int i = tid; i < CC * TPAD; i += 256) {
    int c = i / TPAD, t = i % TPAD;
    float val = 0.f;
    if (t < TT) val = x[((size_t)(n * CC + c) * TT + t) * VV + v] * scl[c] + shf[c];
    Bt[i] = val;
  }
  __syncthreads();

  const int lane  = tid & 31;
  const int wid   = tid >> 5;
  const int row16 = lane & 15;
  const int koff  = (lane < 16) ? 0 : 2;  // A/B K-half per lane group

  for (int mt = wid; mt < QO / 16; mt += 8) {
    const int m0 = mt * 16;
    v2f afrag[32];
#pragma unroll
    for (int kk = 0; kk < 32; ++kk)
      afrag[kk] = *(const v2f*)(qkv_w + (size_t)(m0 + row16) * CC + kk * 4 + koff);
    for (int t0 = 0; t0 < TT; t0 += 16) {
      v8f acc = {};
#pragma unroll
      for (int kk = 0; kk < 32; ++kk) {
        v2f bb;
        bb.x = Bt[(kk * 4 + koff) * TPAD + t0 + row16];
        bb.y = Bt[(kk * 4 + koff + 1) * TPAD + t0 + row16];
        acc = wmma4(afrag[kk], bb, acc);
      }
      const int t = t0 + row16;   // D: col N = lane%16
      if (t < TT) {
#pragma unroll
        for (int j = 0; j < 8; ++j) {
          int o = m0 + j + ((lane >= 16) ? 8 : 0);  // D: row M = j + 8*(lane>=16)
          float r = acc[j] + qkv_b[o];
          if (o < 32) r *= 0.5f;  // fold q scale DKH^-0.5
          qkv_out[(size_t)(b * QO + o) * TT + t] = r;
        }
      }
    }
  }
}

// ---------------------------------------------------------------------------
// Kernel 2: fused attention for one (b, h).
//   Staging: pure copies via global_load_async_to_lds_b32 (ASYNCcnt path).
//   Phase A: logits = q^T k  (K=4 -> one WMMA per 16x16 tile)
//   Phase B: rel = q^T key_rel^T, scatter-added with skew s = m + t - (T-1)
//   Wave-parallel softmax in LDS, then Phase C: out = w * val^T (38 WMMAs).
// Entire 160x160 attention matrix lives in LDS (~100 KB of 320 KB WGP LDS).
// ---------------------------------------------------------------------------
extern "C" __global__ __launch_bounds__(256)
void k_attn(const float* __restrict__ qkv, const float* __restrict__ key_rel,
            float* __restrict__ aout) {
  extern __shared__ float sm[];
  float* qf = sm;                 // LSTR*4, layout [t][d] (q pre-scaled in k_qkv)
  float* kf = qf + LSTR * 4;      // LSTR*4, layout [s][d]
  float* vf = kf + LSTR * 4;      // DVH*TPAD, layout [d][s]
  float* lg = vf + DVHH * TPAD;   // LSTR*LSTR
  const int bh = blockIdx.x;
  const int b = bh >> 3, h = bh & 7;
  const int tid = threadIdx.x;
  const int lane = tid & 31;
  const int wid  = tid >> 5;

  const float* qbase = qkv + (size_t)(b * QO + h * DKHH) * TT;
  const float* kbase = qkv + (size_t)(b * QO + 32 + h * DKHH) * TT;
  const float* vbase = qkv + (size_t)(b * QO + 64 + h * DVHH) * TT;

  // Async staging: global -> LDS without touching VGPR data path.
  for (int i = tid; i < DKHH * TT; i += 256) {       // q, k: [t][d] transposed
    int d = i / TT, t = i % TT;                      // consecutive tid -> consecutive t
    async_g2l_b32(lds_off(qf + t * 4 + d), qbase + d * TT + t);
    async_g2l_b32(lds_off(kf + t * 4 + d), kbase + d * TT + t);
  }
  for (int i = tid; i < DVHH * TT; i += 256) {       // val: [d][s]
    int d = i / TT, s = i % TT;
    async_g2l_b32(lds_off(vf + d * TPAD + s), vbase + d * TT + s);
  }
  // Zero padding regions (async copies can't synthesize zeros).
  if (tid < (LSTR - TT) * DKHH) {                    // q/k rows t in [150,160)
    qf[TT * DKHH + tid] = 0.f;
    kf[TT * DKHH + tid] = 0.f;
  }
  if (tid < DVHH * (TPAD - TT)) {                    // val cols s in [150,152)
    int d = tid / (TPAD - TT), s = TT + tid % (TPAD - TT);
    vf[d * TPAD + s] = 0.f;
  }
  wait_async0();
  __syncthreads();

  const int row16 = lane & 15;
  const int koff  = (lane < 16) ? 0 : 2;

  // Phase A: q^T k -> logits tiles (rows/cols >=150 come out exactly 0)
  for (int tile = wid; tile < 100; tile += 8) {
    int t0 = (tile / 10) * 16, s0 = (tile % 10) * 16;
    v2f a  = *(const v2f*)(qf + (t0 + row16) * 4 + koff);
    v2f bb = *(const v2f*)(kf + (s0 + row16) * 4 + koff);
    v8f c = {};
    c = wmma4(a, bb, c);
#pragma unroll
    for (int j = 0; j < 8; ++j) {
      int t = t0 + j + ((lane >= 16) ? 8 : 0);
      lg[t * LSTR + s0 + row16] = c[j];
    }
  }
  __syncthreads();

  // Phase B: rel tiles + skewed scatter-add (each (t,m) -> unique (t,s))
  for (int tile = wid; tile < 190; tile += 8) {
    int t0 = (tile / 19) * 16, m0 = (tile % 19) * 16;
    v2f a = *(const v2f*)(qf + (t0 + row16) * 4 + koff);
    v2f bb;
    int m = m0 + row16;
    if (m < 2 * TT - 1) {
      bb = *(const v2f*)(key_rel + (size_t)m * 4 + koff);
    } else {
      bb.x = 0.f; bb.y = 0.f;
    }
    v8f c = {};
    c = wmma4(a, bb, c);
#pragma unroll
    for (int j = 0; j < 8; ++j) {
      int t = t0 + j + ((lane >= 16) ? 8 : 0);
      int s = m + t - (TT - 1);
      if (t < TT && s >= 0 && s < TT) lg[t * LSTR + s] += c[j];
    }
  }
  __syncthreads();

  // Wave-parallel softmax: one wave32 per row, shfl_xor reductions.
  for (int r = wid; r < TT; r += 8) {
    float* rowp = lg + r * LSTR;
    float mx = -3.4e38f;
    for (int s = lane; s < TT; s += 32) mx = fmaxf(mx, rowp[s]);
#pragma unroll
    for (int off = 16; off; off >>= 1) mx = fmaxf(mx, __shfl_xor(mx, off, 32));
    float sum = 0.f;
    for (int s = lane; s < TT; s += 32) {
      float e = __expf(rowp[s] - mx);
      rowp[s] = e;
      sum += e;
    }
#pragma unroll
    for (int off = 16; off; off >>= 1) sum += __shfl_xor(sum, off, 32);
    float inv = 1.f / sum;
    for (int s = lane; s < TT; s += 32) rowp[s] *= inv;
  }
  __syncthreads();

  // Phase C: out[t][d] = sum_s w[t][s] * val[d][s]
  float* obase = aout + (size_t)(b * CC + h * DVHH) * TT;
  for (int tt = wid; tt < 10; tt += 8) {
    int t0 = tt * 16;
    v8f acc = {};
#pragma unroll
    for (int kk = 0; kk < 38; ++kk) {
      int k0 = kk * 4;
      v2f a  = *(const v2f*)(lg + (t0 + row16) * LSTR + k0 + koff);
      v2f bb = *(const v2f*)(vf + row16 * TPAD + k0 + koff);
      acc = wmma4(a, bb, acc);
    }
#pragma unroll
    for (int j = 0; j < 8; ++j) {
      int t = t0 + j + ((lane >= 16) ? 8 : 0);
      if (t < TT) obase[row16 * TT + t] = acc[j];  // channel = h*16 + d(row16)
    }
  }
}

// ---------------------------------------------------------------------------
// Kernel 3: output projection (128x128 GEMM) fused with +attn_b, residual x,
// BatchNorm (folded scale/shift) and ReLU. One block per b. B-panel staged
// with async global->LDS copies.
// ---------------------------------------------------------------------------
extern "C" __global__ __launch_bounds__(256)
void k_outproj(const float* __restrict__ aout, const float* __restrict__ attn_w,
               const float* __restrict__ attn_b, const float* __restrict__ x,
               const float* __restrict__ bg, const float* __restrict__ bbeta,
               const float* __restrict__ bm, const float* __restrict__ bv,
               float* __restrict__ out) {
  extern __shared__ float sm[];
  float* Bt  = sm;              // CC * TPAD
  float* scl = Bt + CC * TPAD;  // CC
  float* shf = scl + CC;        // CC
  float* ab  = shf + CC;        // CC
  const int b = blockIdx.x;
  const int n = b / VV, v = b % VV;
  const int tid = threadIdx.x;

  if (tid < CC) {
    float s = bg[tid] * rsqrtf(bv[tid] + 1e-5f);
    scl[tid] = s;
    shf[tid] = bbeta[tid] - bm[tid] * s;
    ab[tid] = attn_b[tid];
  }
  const float* abase = aout + (size_t)b * CC * TT;
  for (int i = tid; i < CC * TT; i += 256) {
    int c = i / TT, t = i % TT;
    async_g2l_b32(lds_off(Bt + c * TPAD + t), abase + i);
  }
  if (tid < CC * (TPAD - TT)) {                      // cols t in [150,152)
    int c = tid / (TPAD - TT), t = TT + tid % (TPAD - TT);
    Bt[c * TPAD + t] = 0.f;
  }
  wait_async0();
  __syncthreads();

  const int lane  = tid & 31;
  const int wid   = tid >> 5;
  const int row16 = lane & 15;
  const int koff  = (lane < 16) ? 0 : 2;
  const int m0    = wid * 16;   // 8 waves <-> 8 m-tiles of 128

  v2f afrag[32];
#pragma unroll
  for (int kk = 0; kk < 32; ++kk)
    afrag[kk] = *(const v2f*)(attn_w + (size_t)(m0 + row16) * CC + kk * 4 + koff);

  for (int t0 = 0; t0 < TT; t0 += 16) {
    v8f acc = {};
#pragma unroll
    for (int kk = 0; kk < 32; ++kk) {
      v2f bbv;
      bbv.x = Bt[(kk * 4 + koff) * TPAD + t0 + row16];
      bbv.y = Bt[(kk * 4 + koff + 1) * TPAD + t0 + row16];
      acc = wmma4(afrag[kk], bbv, acc);
    }
    const int t = t0 + row16;
    if (t < TT) {
#pragma unroll
      for (int j = 0; j < 8; ++j) {
        int o = m0 + j + ((lane >= 16) ? 8 : 0);
        size_t xi = ((size_t)(n * CC + o) * TT + t) * VV + v;
        float r = acc[j] + ab[o] + x[xi];
        r = r * scl[o] + shf[o];
        out[xi] = fmaxf(r, 0.f);
      }
    }
  }
}

// ---------------------------------------------------------------------------
extern "C" void kernel_launch(void* const* d_in, const int* in_sizes, int n_in,
                              void* d_out, int out_size, void* d_ws, size_t ws_size,
                              hipStream_t stream) {
  const float* x       = (const float*)d_in[0];
  const float* qkv_w   = (const float*)d_in[1];
  const float* qkv_b   = (const float*)d_in[2];
  const float* key_rel = (const float*)d_in[3];
  const float* attn_w  = (const float*)d_in[4];
  const float* attn_b  = (const float*)d_in[5];
  const float* dbn_g   = (const float*)d_in[6];
  const float* dbn_b   = (const float*)d_in[7];
  const float* dbn_m   = (const float*)d_in[8];
  const float* dbn_v   = (const float*)d_in[9];
  const float* bn_g    = (const float*)d_in[10];
  const float* bn_b    = (const float*)d_in[11];
  const float* bn_m    = (const float*)d_in[12];
  const float* bn_v    = (const float*)d_in[13];

  float* qkv_ws  = (float*)d_ws;                          // 400*192*150 f32
  float* aout_ws = qkv_ws + (size_t)400 * QO * TT;        // 400*128*150 f32
  float* out     = (float*)d_out;

  size_t sm1 = (size_t)(CC * TPAD + 2 * CC) * sizeof(float);
  size_t sm2 = (size_t)(LSTR * 4 * 2 + DVHH * TPAD + LSTR * LSTR) * sizeof(float);
  size_t sm3 = (size_t)(CC * TPAD + 3 * CC) * sizeof(float);

  k_qkv<<<dim3(NB * VV), dim3(256), sm1, stream>>>(x, qkv_w, qkv_b, dbn_g, dbn_b,
                                                   dbn_m, dbn_v, qkv_ws);
  k_attn<<<dim3(NB * VV * NHH), dim3(256), sm2, stream>>>(qkv_ws, key_rel, aout_ws);
  k_outproj<<<dim3(NB * VV), dim3(256), sm3, stream>>>(aout_ws, attn_w, attn_b, x,
                                                       bn_g, bn_b, bn_m, bn_v, out);
}